// BlockwiseQuantizationOptim_300647711204
// MI455X (gfx1250) — compile-verified
//
#include <hip/hip_runtime.h>
#include <cstdint>
#include <cstddef>

#ifndef __has_builtin
#define __has_builtin(x) 0
#endif

// CDNA5 async global->LDS copy path (gfx1250). Gated so the file still
// compiles (with a synchronous fallback) if the toolchain lacks the builtins.
#if defined(__AMDGCN__) &&                                            \
    __has_builtin(__builtin_amdgcn_global_load_async_to_lds_b128) &&  \
    __has_builtin(__builtin_amdgcn_s_wait_asynccnt)
#define USE_ASYNC_COPY 1
#else
#define USE_ASYNC_COPY 0
#endif

namespace {
constexpr int   kDim       = 4096;
constexpr int   kBlk       = 128;                 // quant block is 128x128
constexpr int   kChunkRows = 32;                  // rows staged per LDS chunk
constexpr int   kNChunks   = kBlk / kChunkRows;   // 4
constexpr int   kLev       = 16;
constexpr float kEps       = 1e-6f;
// softmax temperature in log2 units: (100/15)/ln(2)
constexpr float kG2  = 9.6179669f;
constexpr float kLn2 = 0.69314718f;
}  // namespace

// The async-load builtin takes (global int4*, lds int4*, imm offset, imm cpol)
// per the compile diagnostic (param type: 'int __vector(4) *').
typedef int v4i_t __attribute__((ext_vector_type(4)));
typedef __attribute__((address_space(1))) v4i_t gv4i_t;
typedef __attribute__((address_space(3))) v4i_t lv4i_t;

__device__ __forceinline__ float fast_exp2(float x) {
#if defined(__AMDGCN__) && __has_builtin(__builtin_amdgcn_exp2f)
  return __builtin_amdgcn_exp2f(x);          // v_exp_f32
#else
  return exp2f(x);
#endif
}
__device__ __forceinline__ float fast_log2(float x) {
#if defined(__AMDGCN__) && __has_builtin(__builtin_amdgcn_logf)
  return __builtin_amdgcn_logf(x);           // v_log_f32
#else
  return log2f(x);
#endif
}
__device__ __forceinline__ float fast_rcp(float x) {
#if defined(__AMDGCN__) && __has_builtin(__builtin_amdgcn_rcpf)
  return __builtin_amdgcn_rcpf(x);           // v_rcp_f32
#else
  return 1.0f / x;
#endif
}

// Stage one 32x128 fp32 chunk (16 KB) of the weight tile into LDS.
// 256 threads * 4 x b128 = 1024 16-byte vectors. Fully coalesced.
__device__ __forceinline__ void issue_chunk_loads(const float* __restrict__ w,
                                                  int row_base, int col0,
                                                  float* lds_dst, int tid) {
#pragma unroll
  for (int m = 0; m < 4; ++m) {
    const int v  = tid + 256 * m;   // vector index within chunk
    const int r  = v >> 5;          // 32 vectors per 128-float row
    const int cf = (v & 31) << 2;   // float column within row
    const float* g = w + (size_t)(row_base + r) * kDim + col0 + cf;
#if USE_ASYNC_COPY
    __builtin_amdgcn_global_load_async_to_lds_b128(
        (gv4i_t*)(uintptr_t)g,
        (lv4i_t*)(uint32_t)(uintptr_t)(lds_dst + v * 4),
        0, 0);
#else
    *(float4*)(lds_dst + v * 4) = *(const float4*)g;
#endif
  }
}

__global__ void __launch_bounds__(256)
bq455_quant_kernel(const float* __restrict__ w,
                   const float* __restrict__ wmin_in,
                   const float* __restrict__ wmax_in,
                   float* __restrict__ out,
                   float* __restrict__ ent_partial) {
  __shared__ float s_buf[2][kChunkRows * kBlk];  // 2 x 16 KB double buffer
  __shared__ float s_hist[kLev * kBlk];          // 8 KB  [level][col]
  __shared__ float s_red[256];                   // 1 KB reduction scratch

  const int tid  = threadIdx.x;
  const int bid  = blockIdx.x;           // 1024 blocks: br*32 + bc
  const int br   = bid >> 5;
  const int bc   = bid & 31;
  const int row0 = br * kBlk;
  const int col0 = bc * kBlk;
  const int col  = tid & (kBlk - 1);     // lane -> consecutive columns
  const int half = tid >> 7;             // 2 threads per column (64 rows each)

  // Adjusted block min/max (matches reference order of ops).
  float wmn = wmin_in[bid];
  float wmx = wmax_in[bid];
  wmn = fminf(wmn, wmx - kEps);
  wmx = fmaxf(wmx, wmn + kEps);
  const float range     = wmx - wmn;
  const float inv_scale = 1.0f / (range + kEps);

#pragma unroll
  for (int i = 0; i < (kLev * kBlk) / 256; ++i)  // zero 8 entries each
    s_hist[tid + 256 * i] = 0.0f;

  issue_chunk_loads(w, row0, col0, &s_buf[0][0], tid);

  float acc[kLev];
#pragma unroll
  for (int k = 0; k < kLev; ++k) acc[k] = 0.0f;

  for (int c = 0; c < kNChunks; ++c) {
    if (c + 1 < kNChunks) {
      issue_chunk_loads(w, row0 + (c + 1) * kChunkRows, col0,
                        &s_buf[(c + 1) & 1][0], tid);
#if USE_ASYNC_COPY
      __builtin_amdgcn_s_wait_asynccnt(4);  // chunk c done, next 4 in flight
#endif
    } else {
#if USE_ASYNC_COPY
      __builtin_amdgcn_s_wait_asynccnt(0);
#endif
    }
    __syncthreads();

    const float* cb = &s_buf[c & 1][0];
    for (int i = 0; i < kChunkRows / 2; ++i) {      // 16 rows per thread
      const int   lrow = half * (kChunkRows / 2) + i;
      const float x    = cb[lrow * kBlk + col];     // conflict-free ds_load
      const float t15  = (x - wmn) * inv_scale * 15.0f;

      // softmax over 16 levels: u_k = exp(-T*|w_norm - k/15|), done in log2.
      float u[kLev];
      float S = 0.0f, Wm = 0.0f;
#pragma unroll
      for (int k = 0; k < kLev; ++k) {
        const float d = t15 - (float)k;
        u[k] = fast_exp2(-kG2 * fabsf(d));
        S += u[k];
        Wm = fmaf(u[k], (float)k, Wm);
      }
      const float invS = fast_rcp(S);
#pragma unroll
      for (int k = 0; k < kLev; ++k)                 // per-thread bin masses
        acc[k] = fmaf(u[k], invS, acc[k]);

      const float wq = Wm * invS * (1.0f / 15.0f);   // sum(p_k * level_k)
      out[(size_t)(row0 + c * kChunkRows + lrow) * kDim + col0 + col] =
          fmaf(wq, range, wmn);                      // dequantize
    }
    __syncthreads();  // chunk consumed before its buffer is refilled
  }

  // Fold per-thread level masses into the per-(level,col) histogram.
  // [level][col] layout -> lanes hit consecutive LDS banks; 2-way commutative
  // ds_add_f32 per location (bitwise deterministic).
#pragma unroll
  for (int k = 0; k < kLev; ++k)
    atomicAdd(&s_hist[k * kBlk + col], acc[k]);
  __syncthreads();

  // Total bin mass for this block (fixed-order tree reduction).
  float part = 0.0f;
#pragma unroll
  for (int i = 0; i < 8; ++i) part += s_hist[tid * 8 + i];
  s_red[tid] = part;
  __syncthreads();
  for (int st = 128; st > 0; st >>= 1) {
    if (tid < st) s_red[tid] += s_red[tid + st];
    __syncthreads();
  }
  const float inv_tot = 1.0f / (s_red[0] + kEps);
  __syncthreads();

  // Entropy contribution: -sum p*ln(p + eps)
  float e = 0.0f;
#pragma unroll
  for (int i = 0; i < 8; ++i) {
    const float p = s_hist[tid * 8 + i] * inv_tot;
    e = fmaf(p, fast_log2(p + kEps) * kLn2, e);
  }
  s_red[tid] = e;
  __syncthreads();
  for (int st = 128; st > 0; st >>= 1) {
    if (tid < st) s_red[tid] += s_red[tid + st];
    __syncthreads();
  }
  if (tid == 0) ent_partial[bid] = -s_red[0];
}

// Deterministic final sum of the 1024 per-block entropies.
__global__ void __launch_bounds__(256)
bq455_ent_reduce_kernel(const float* __restrict__ ent_partial,
                        float* __restrict__ out_ent) {
  __shared__ float s_red[256];
  const int tid = threadIdx.x;
  float s = 0.0f;
#pragma unroll
  for (int i = 0; i < 4; ++i) s += ent_partial[tid + 256 * i];
  s_red[tid] = s;
  __syncthreads();
  for (int st = 128; st > 0; st >>= 1) {
    if (tid < st) s_red[tid] += s_red[tid + st];
    __syncthreads();
  }
  if (tid == 0) *out_ent = s_red[0];
}

extern "C" void kernel_launch(void* const* d_in, const int* in_sizes, int n_in,
                              void* d_out, int out_size, void* d_ws,
                              size_t ws_size, hipStream_t stream) {
  (void)in_sizes; (void)n_in; (void)out_size; (void)ws_size;
  const float* weight = (const float*)d_in[0];   // 4096*4096 fp32
  const float* w_min  = (const float*)d_in[1];   // 1024 fp32
  const float* w_max  = (const float*)d_in[2];   // 1024 fp32
  float* out = (float*)d_out;                    // 4096*4096 + 1 fp32
  float* ws  = (float*)d_ws;                     // >= 1024 floats scratch

  bq455_quant_kernel<<<1024, 256, 0, stream>>>(weight, w_min, w_max, out, ws);
  bq455_ent_reduce_kernel<<<1, 256, 0, stream>>>(ws,
                                                 out + (size_t)4096 * 4096);
}